// ChamferDistKDTree_7421703487936
// MI455X (gfx1250) — compile-verified
//
#include <hip/hip_runtime.h>

typedef float v2f __attribute__((ext_vector_type(2)));
typedef float v8f __attribute__((ext_vector_type(8)));

#define TILE_ROWS 128          // src rows per block (8 waves x 16 rows)
#define WAVES     8
#define CHUNK     2048         // tar points staged in LDS per iteration
#define PAD2      16           // float2 pad -> region 1 lands on banks 32..63
#define BLOCK     256

// For each of TILE_ROWS query points, find min squared distance to all nB
// reference points; emit  sum_rows sqrt(max(min_d2, 0))  as one partial.
//
// Math: with A' rows = (-2x, -2y, -2z, 1) and B cols = (x, y, z, |b|^2),
//       WMMA_F32_16X16X4 gives D = |b|^2 - 2 a.b  (C = inline 0).
//       min_col D + |a|^2 = min squared distance for that row.
__global__ __launch_bounds__(BLOCK) void chamfer_nn_kernel(
    const float* __restrict__ Apts,   // [B, nA, 3] query cloud
    const float* __restrict__ Bpts,   // [B, nB, 3] reference cloud
    float* __restrict__ partials,     // one partial sum per block
    int nA, int nB, int tilesPerBatch, int dirOffset)
{
    // fragment-layout tar chunk: [0..CHUNK) = (x,y), [CHUNK+PAD2..) = (z,|b|^2)
    __shared__ v2f    fragbuf[2 * CHUNK + PAD2];
    __shared__ float4 srcbuf[TILE_ROWS];          // (x, y, z, |a|^2)
    __shared__ float  wavesum[WAVES * 2];

    const int tid  = threadIdx.x;
    const int wave = tid >> 5;
    const int lane = tid & 31;
    const int half = lane >> 4;   // 0: K=0..1 / rows 0..7 ; 1: K=2..3 / rows 8..15
    const int l16  = lane & 15;

    const int b    = blockIdx.x / tilesPerBatch;
    const int tile = blockIdx.x % tilesPerBatch;

    const float* __restrict__ Abase = Apts + ((size_t)b * nA + (size_t)tile * TILE_ROWS) * 3;
    const float* __restrict__ Bbase = Bpts + (size_t)b * nB * 3;

    // ---- stage this block's 128 src rows (with norms) into LDS ----
    if (tid < TILE_ROWS) {
        float x = Abase[tid * 3 + 0];
        float y = Abase[tid * 3 + 1];
        float z = Abase[tid * 3 + 2];
        srcbuf[tid] = make_float4(x, y, z, x * x + y * y + z * z);
    }
    __syncthreads();

    // ---- A fragment: A' = (-2x,-2y | -2z, 1)  (K=3 element = 1 pulls |b|^2 in) ----
    const int rowbase = wave * 16;
    float4 sp = srcbuf[rowbase + l16];
    v2f afrag;
    if (half == 0) { afrag.x = -2.0f * sp.x; afrag.y = -2.0f * sp.y; }
    else           { afrag.x = -2.0f * sp.z; afrag.y = 1.0f; }

    // |a|^2 for the 8 rows this lane's D elements correspond to
    float snorm[8];
    #pragma unroll
    for (int j = 0; j < 8; ++j)
        snorm[j] = srcbuf[rowbase + half * 8 + j].w;

    // per-lane fragment base: region picked once, loop uses immediate offsets
    const v2f* __restrict__ fragbase = &fragbuf[half * (CHUNK + PAD2) + l16];

    v8f rm0, rm1;
    #pragma unroll
    for (int j = 0; j < 8; ++j) { rm0[j] = 3.0e38f; rm1[j] = 3.0e38f; }

    // ---- sweep reference cloud in LDS-resident chunks ----
    for (int c0 = 0; c0 < nB; c0 += CHUNK) {
        __syncthreads();  // previous chunk's readers done before overwrite
        #pragma unroll
        for (int k = 0; k < CHUNK / BLOCK; ++k) {
            int p = k * BLOCK + tid;
            const float* q = Bbase + (size_t)(c0 + p) * 3;
            float x = q[0], y = q[1], z = q[2];
            v2f xy; xy.x = x; xy.y = y;
            v2f zn; zn.x = z; zn.y = x * x + y * y + z * z;
            fragbuf[p] = xy;
            fragbuf[CHUNK + PAD2 + p] = zn;
        }
        __syncthreads();

        // 64 reference columns per iteration: 4 WMMAs, 2 independent accumulators
        for (int cc = 0; cc < CHUNK; cc += 64) {
            v2f b0 = fragbase[cc +  0];
            v2f b1 = fragbase[cc + 16];
            v2f b2 = fragbase[cc + 32];
            v2f b3 = fragbase[cc + 48];
            v8f cz = {};   // inline 0 for SRC2
            v8f d0 = __builtin_amdgcn_wmma_f32_16x16x4_f32(false, afrag, false, b0, (short)0, cz, false, false);
            v8f d1 = __builtin_amdgcn_wmma_f32_16x16x4_f32(false, afrag, false, b1, (short)0, cz, false, false);
            v8f d2 = __builtin_amdgcn_wmma_f32_16x16x4_f32(false, afrag, false, b2, (short)0, cz, false, false);
            v8f d3 = __builtin_amdgcn_wmma_f32_16x16x4_f32(false, afrag, false, b3, (short)0, cz, false, false);
            #pragma unroll
            for (int j = 0; j < 8; ++j) {
                rm0[j] = fminf(rm0[j], fminf(d0[j], d1[j]));   // -> v_min3
                rm1[j] = fminf(rm1[j], fminf(d2[j], d3[j]));   // -> v_min3
            }
        }
    }

    v8f runmin;
    #pragma unroll
    for (int j = 0; j < 8; ++j) runmin[j] = fminf(rm0[j], rm1[j]);

    // ---- min across the 16 lanes of each half (columns) ----
    #pragma unroll
    for (int m = 1; m <= 8; m <<= 1) {
        #pragma unroll
        for (int j = 0; j < 8; ++j)
            runmin[j] = fminf(runmin[j], __shfl_xor(runmin[j], m, 32));
    }

    // ---- add |a|^2, clamp, sqrt, sum the 8 rows ----
    float s = 0.0f;
    #pragma unroll
    for (int j = 0; j < 8; ++j) {
        float d2 = runmin[j] + snorm[j];
        d2 = d2 > 0.0f ? d2 : 0.0f;
        s += sqrtf(d2);
    }
    if (l16 == 0) wavesum[wave * 2 + half] = s;  // rows 0-7 (half 0), 8-15 (half 1)
    __syncthreads();

    if (tid == 0) {
        float tot = 0.0f;
        #pragma unroll
        for (int i = 0; i < WAVES * 2; ++i) tot += wavesum[i];
        partials[dirOffset + blockIdx.x] = tot;
    }
}

// Deterministic two-stage reduction: sum partials per batch, combine directions.
__global__ void chamfer_finalize(const float* __restrict__ partials,
                                 float* __restrict__ out,
                                 int tilesA, int tilesB, int nA, int nB, int batches)
{
    int b = threadIdx.x;
    if (b < batches) {
        float s0 = 0.0f, s1 = 0.0f;
        for (int t = 0; t < tilesA; ++t) s0 += partials[b * tilesA + t];
        int off = batches * tilesA;
        for (int t = 0; t < tilesB; ++t) s1 += partials[off + b * tilesB + t];
        out[b] = 0.5f * (s0 / (float)nA + s1 / (float)nB);
    }
}

extern "C" void kernel_launch(void* const* d_in, const int* in_sizes, int n_in,
                              void* d_out, int out_size, void* d_ws, size_t ws_size,
                              hipStream_t stream) {
    const float* src = (const float*)d_in[0];  // [B, M, 3]
    const float* tar = (const float*)d_in[1];  // [B, N, 3]
    float* out = (float*)d_out;                // [B]
    float* ws  = (float*)d_ws;

    const int B = out_size;                    // reference output is [B]
    const int M = in_sizes[0] / (B * 3);
    const int N = in_sizes[1] / (B * 3);
    const int tilesA = M / TILE_ROWS;
    const int tilesB = N / TILE_ROWS;

    // accuracy: src -> tar   (partials ws[0 .. B*tilesA))
    chamfer_nn_kernel<<<dim3(B * tilesA), dim3(BLOCK), 0, stream>>>(
        src, tar, ws, M, N, tilesA, 0);
    // completeness: tar -> src  (partials ws[B*tilesA .. B*tilesA + B*tilesB))
    chamfer_nn_kernel<<<dim3(B * tilesB), dim3(BLOCK), 0, stream>>>(
        tar, src, ws, N, M, tilesB, B * tilesA);

    chamfer_finalize<<<1, 32, 0, stream>>>(ws, out, tilesA, tilesB, M, N, B);
}